// Morphological_38250978738448
// MI455X (gfx1250) — compile-verified
//
#include <hip/hip_runtime.h>
#include <math.h>

// ---------------------------------------------------------------------------
// Tropical (max-plus) GEMM: out[b,j] = max_i (x[b,i] + w[i,j])
// B=2048, I=1024, J=1024, f32.
// Compute-bound on VALU (no WMMA mapping exists for the max-plus semiring).
// CDNA5 path used: async global->LDS DMA (ASYNCcnt) double-buffered pipeline
// + ds_load_b128 register-blocked tiles (16 VALU ops per LDS b128 load).
// ---------------------------------------------------------------------------

#ifndef __has_builtin
#define __has_builtin(x) 0
#endif

#if __has_builtin(__builtin_amdgcn_global_load_async_to_lds_b128)
#define USE_ASYNC_LDS 1
#else
#define USE_ASYNC_LDS 0
#endif

#define GLOBAL_AS __attribute__((address_space(1)))
#define LDS_AS __attribute__((address_space(3)))

typedef int v4i __attribute__((ext_vector_type(4)));

static __device__ __forceinline__ void wait_asynccnt0() {
#if __has_builtin(__builtin_amdgcn_s_wait_asynccnt)
  __builtin_amdgcn_s_wait_asynccnt(0);
#else
  asm volatile("s_wait_asynccnt 0" ::: "memory");
#endif
}

namespace {
constexpr int Bdim = 2048;
constexpr int Idim = 1024;
constexpr int Jdim = 1024;

constexpr int TB = 64;          // output tile rows (b)
constexpr int TJ = 64;          // output tile cols (j)
constexpr int TK = 64;          // reduction chunk
constexpr int NKT = Idim / TK;  // 16 k-tiles
constexpr int NTHREADS = 256;   // 8 wave32
}  // namespace

__global__ __launch_bounds__(NTHREADS) void maxplus_tiled_kernel(
    const float* __restrict__ x, const float* __restrict__ w,
    float* __restrict__ out) {
  // Double-buffered tiles: 2 * (64*64 + 64*64) * 4B = 64 KB (of 320 KB WGP LDS)
  __shared__ float xs[2][TB * TK];  // [row][k], row-major
  __shared__ float ws[2][TK * TJ];  // [k][col], row-major

  const int tid = threadIdx.x;
  const int b0 = blockIdx.y * TB;
  const int j0 = blockIdx.x * TJ;

  const int tx = tid & 15;  // 16 col-groups
  const int ty = tid >> 4;  // 16 row-groups
  const int c0 = tx << 2;   // this thread's 4 cols within tile
  const int r0 = ty << 2;   // this thread's 4 rows within tile

  // Issue one 64x64 f32 tile copy (global -> LDS) as 16B granules.
  // 1024 granules / 256 threads = 4 async b128 ops per thread.
  auto issue_x = [&](int buf, int k0) {
#pragma unroll
    for (int i = 0; i < 4; ++i) {
      const int g = tid + i * NTHREADS;
      const int row = g >> 4;
      const int c4 = (g & 15) << 2;
      const float* src = x + (size_t)(b0 + row) * Idim + (k0 + c4);
      float* dst = &xs[buf][row * TK + c4];
#if USE_ASYNC_LDS
      __builtin_amdgcn_global_load_async_to_lds_b128(
          (GLOBAL_AS v4i*)src, (LDS_AS v4i*)dst, 0, 0);
#else
      *(float4*)dst = *(const float4*)src;
#endif
    }
  };
  auto issue_w = [&](int buf, int k0) {
#pragma unroll
    for (int i = 0; i < 4; ++i) {
      const int g = tid + i * NTHREADS;
      const int krow = g >> 4;
      const int c4 = (g & 15) << 2;
      const float* src = w + (size_t)(k0 + krow) * Jdim + (j0 + c4);
      float* dst = &ws[buf][krow * TJ + c4];
#if USE_ASYNC_LDS
      __builtin_amdgcn_global_load_async_to_lds_b128(
          (GLOBAL_AS v4i*)src, (LDS_AS v4i*)dst, 0, 0);
#else
      *(float4*)dst = *(const float4*)src;
#endif
    }
  };

  float acc[4][4];
#pragma unroll
  for (int r = 0; r < 4; ++r)
#pragma unroll
    for (int c = 0; c < 4; ++c) acc[r][c] = -INFINITY;

  // Register-blocked tropical inner product on one LDS buffer.
  // Per 4 k-steps: 8x ds_load_b128 feed 128 VALU ops (64 add + 64 max).
  auto compute = [&](int buf) {
    const float* xb = &xs[buf][0];
    const float* wb = &ws[buf][0];
#pragma unroll
    for (int kk = 0; kk < TK; kk += 4) {
      float xv[4][4];
      float wv[4][4];
#pragma unroll
      for (int r = 0; r < 4; ++r) {
        const float4 v = *(const float4*)(xb + (r0 + r) * TK + kk);
        xv[r][0] = v.x; xv[r][1] = v.y; xv[r][2] = v.z; xv[r][3] = v.w;
      }
#pragma unroll
      for (int t = 0; t < 4; ++t) {
        const float4 v = *(const float4*)(wb + (kk + t) * TJ + c0);
        wv[t][0] = v.x; wv[t][1] = v.y; wv[t][2] = v.z; wv[t][3] = v.w;
      }
#pragma unroll
      for (int t = 0; t < 4; ++t)
#pragma unroll
        for (int r = 0; r < 4; ++r)
#pragma unroll
          for (int c = 0; c < 4; ++c)
            acc[r][c] = fmaxf(acc[r][c], xv[r][t] + wv[t][c]);
    }
  };

#if USE_ASYNC_LDS
  // Double-buffered async pipeline.
  issue_x(0, 0);
  issue_w(0, 0);
#pragma unroll 1
  for (int kt = 0; kt < NKT; ++kt) {
    wait_asynccnt0();  // my wave's DMA for buffer kt&1 has landed in LDS
    __syncthreads();   // everyone's DMA has landed; everyone done reading ^1
    if (kt + 1 < NKT) {
      const int nbuf = (kt + 1) & 1;
      issue_x(nbuf, (kt + 1) * TK);
      issue_w(nbuf, (kt + 1) * TK);
    }
    compute(kt & 1);
  }
#else
  // Fallback: synchronous single-buffer staging.
#pragma unroll 1
  for (int kt = 0; kt < NKT; ++kt) {
    __syncthreads();
    issue_x(0, kt * TK);
    issue_w(0, kt * TK);
    __syncthreads();
    compute(0);
  }
#endif

  // Write the 4x4 block with b128 stores.
#pragma unroll
  for (int r = 0; r < 4; ++r) {
    float4 v;
    v.x = acc[r][0]; v.y = acc[r][1]; v.z = acc[r][2]; v.w = acc[r][3];
    *(float4*)(out + (size_t)(b0 + r0 + r) * Jdim + (j0 + c0)) = v;
  }
}

extern "C" void kernel_launch(void* const* d_in, const int* in_sizes, int n_in,
                              void* d_out, int out_size, void* d_ws,
                              size_t ws_size, hipStream_t stream) {
  (void)in_sizes; (void)n_in; (void)out_size; (void)d_ws; (void)ws_size;
  const float* x = (const float*)d_in[0];  // [2048, 1024]
  const float* w = (const float*)d_in[1];  // [1024, 1024]
  float* out = (float*)d_out;              // [2048, 1024]

  dim3 grid(Jdim / TJ, Bdim / TB);  // (16, 32) = 512 workgroups
  dim3 block(NTHREADS);             // 8 wave32
  maxplus_tiled_kernel<<<grid, block, 0, stream>>>(x, w, out);
}